// MyConv_45655502356534
// MI455X (gfx1250) — compile-verified
//
#include <hip/hip_runtime.h>
#include <hip/hip_bf16.h>
#include <math.h>

typedef __attribute__((ext_vector_type(2))) float v2f;
typedef __attribute__((ext_vector_type(8))) float v8f;

#define C_IN   64
#define C_OUT  64
#define K_PAD  72      // 71 padded to multiple of 4
#define W_STR  66      // padded LDS row stride for weights (bank-conflict avoidance)
#define WAVES  4       // waves per block (128 threads)

// ---- order-preserving float -> uint key (monotonic under unsigned compare) ----
__device__ __forceinline__ unsigned f2key(float f) {
    unsigned u = __float_as_uint(f);
    // negative: ~u ; positive: u | 0x80000000  -- single xor with computed mask
    return u ^ (unsigned)(((int)u >> 31) | 0x80000000);
}

// ------------------- kernel 1: init keys to 0 (== "empty segment") -------------------
__global__ void init_keys(unsigned* __restrict__ keys, int total) {
    int i = blockIdx.x * blockDim.x + threadIdx.x;
    if (i < total) keys[i] = 0u;
}

// ------------------- kernel 2: per-edge message GEMM via WMMA + atomic scatter-max ---
__global__ __launch_bounds__(128)
void edge_wmma_kernel(const float* __restrict__ feat,      // [N,64]
                      const float* __restrict__ pos,       // [N,3]
                      const float* __restrict__ Wg,        // [2,71,64]
                      const float* __restrict__ bg,        // [2,64]
                      const int*   __restrict__ eidx,      // [2,E]
                      const int*   __restrict__ etyp,      // [E]
                      unsigned*    __restrict__ keys,      // [N,64] as keys
                      int E, int nTiles, int nIter)
{
    __shared__ __align__(16) float sW[2][K_PAD][W_STR];     // padded weights (k=71 row zeroed)
    __shared__ __align__(16) float sMsg[WAVES][16][K_PAD];  // per-wave 16x72 message tile
    __shared__ __align__(16) int sDst[WAVES][16];
    __shared__ __align__(16) int sTyp[WAVES][16];

    const int tid  = threadIdx.x;
    const int lane = tid & 31;
    const int wid  = tid >> 5;
    const int row  = lane & 15;   // M row (edge within tile) / N col
    const int half = lane >> 4;   // selects K sub-pair for A/B fragments

    // ---- stage both weight matrices into LDS, zero-padding k=71 ----
    for (int i = tid; i < 2 * K_PAD * C_OUT; i += blockDim.x) {
        int t = i / (K_PAD * C_OUT);
        int r = i - t * (K_PAD * C_OUT);
        int k = r / C_OUT;
        int n = r - k * C_OUT;
        sW[t][k][n] = (k < 71) ? Wg[(t * 71 + k) * C_OUT + n] : 0.0f;
    }

    // ---- hoist per-lane bias values (loop-invariant across tiles) ----
    float bias0[4], bias1[4];
    #pragma unroll
    for (int nt = 0; nt < 4; ++nt) {
        const int n = nt * 16 + row;
        bias0[nt] = bg[n];
        bias1[nt] = bg[C_OUT + n];
    }
    __syncthreads();

    const int slots = gridDim.x * WAVES;
    const int slot  = blockIdx.x * WAVES + wid;

    for (int it = 0; it < nIter; ++it) {
        const int tile = slot + it * slots;
        const bool active = (tile < nTiles);   // wave-uniform

        if (active) {
            // ---- build 16x72 message tile in LDS ----
            const int e = tile * 16 + row;
            if (e < E) {
                const int s = eidx[e];
                // features: each half-lane copies 32 floats (8x float4), coalesced
                const float4* fp = (const float4*)(feat + (size_t)s * C_IN + half * 32);
                float4* mp = (float4*)(&sMsg[wid][row][half * 32]);
                #pragma unroll
                for (int j = 0; j < 8; ++j) mp[j] = fp[j];
                if (half == 0) {
                    const int d = eidx[E + e];
                    float px = pos[3 * s + 0], py = pos[3 * s + 1], pz = pos[3 * s + 2];
                    float qx = pos[3 * d + 0], qy = pos[3 * d + 1], qz = pos[3 * d + 2];
                    float dx = qx - px, dy = qy - py, dz = qz - pz;
                    float* mr = &sMsg[wid][row][0];
                    mr[64] = px; mr[65] = py; mr[66] = pz;
                    mr[67] = dx; mr[68] = dy; mr[69] = dz;
                    mr[70] = sqrtf(dx * dx + dy * dy + dz * dz);
                    mr[71] = 0.0f;
                    sDst[wid][row] = d;
                    sTyp[wid][row] = etyp[e];
                }
            } else if (half == 0) {
                float* mr = &sMsg[wid][row][0];
                for (int k = 0; k < K_PAD; ++k) mr[k] = 0.0f;
                sDst[wid][row] = -1;
                sTyp[wid][row] = 0;
            }
        }
        __syncthreads();

        if (active) {
            // ---- pull this lane's 8 output-row descriptors into registers ----
            // C/D layout: lane holds rows m = v + 8*half, v = 0..7
            int4 d_lo = *(const int4*)&sDst[wid][8 * half + 0];
            int4 d_hi = *(const int4*)&sDst[wid][8 * half + 4];
            int4 t_lo = *(const int4*)&sTyp[wid][8 * half + 0];
            int4 t_hi = *(const int4*)&sTyp[wid][8 * half + 4];
            const int dsts[8] = { d_lo.x, d_lo.y, d_lo.z, d_lo.w,
                                  d_hi.x, d_hi.y, d_hi.z, d_hi.w };
            const int typs[8] = { t_lo.x, t_lo.y, t_lo.z, t_lo.w,
                                  t_hi.x, t_hi.y, t_hi.z, t_hi.w };

            // ---- WMMA: D[16x64] = msg[16x72] @ W[t][72x64], both types ----
            v8f acc0[4] = {};   // type 0, four N-tiles
            v8f acc1[4] = {};   // type 1
            const float* msgRow = &sMsg[wid][row][0];
            const int koff = half * 2;

            for (int kk = 0; kk < 18; ++kk) {
                const int k = kk * 4 + koff;
                v2f a = *(const v2f*)&msgRow[k];
                #pragma unroll
                for (int nt = 0; nt < 4; ++nt) {
                    const int n = nt * 16 + row;
                    v2f b0, b1;
                    b0.x = sW[0][k][n];     b0.y = sW[0][k + 1][n];
                    b1.x = sW[1][k][n];     b1.y = sW[1][k + 1][n];
                    acc0[nt] = __builtin_amdgcn_wmma_f32_16x16x4_f32(
                        false, a, false, b0, (short)0, acc0[nt], false, false);
                    acc1[nt] = __builtin_amdgcn_wmma_f32_16x16x4_f32(
                        false, a, false, b1, (short)0, acc1[nt], false, false);
                }
            }

            // ---- epilogue: select per-edge type, add bias, atomic scatter-max ----
            #pragma unroll
            for (int v = 0; v < 8; ++v) {
                const int d = dsts[v];
                if (d >= 0) {
                    const int t = typs[v];
                    unsigned* basep = keys + (size_t)d * C_OUT + row;
                    #pragma unroll
                    for (int nt = 0; nt < 4; ++nt) {
                        float val = t ? (acc1[nt][v] + bias1[nt])
                                      : (acc0[nt][v] + bias0[nt]);
                        atomicMax(basep + nt * 16, f2key(val));
                    }
                }
            }
        }
        __syncthreads();
    }
}

// ------------------- kernel 3: decode keys in place; empty / non-finite -> 0 --------
__global__ void finalize_keys(float* __restrict__ out, int total) {
    int i = blockIdx.x * blockDim.x + threadIdx.x;
    if (i >= total) return;
    unsigned key = __float_as_uint(out[i]);   // buffer currently holds keys
    float r = 0.0f;
    if (key != 0u) {
        unsigned u = (key & 0x80000000u) ? (key ^ 0x80000000u) : ~key;
        // keep only finite values (matches reference isfinite filter)
        if ((u & 0x7F800000u) != 0x7F800000u) r = __uint_as_float(u);
    }
    out[i] = r;
}

extern "C" void kernel_launch(void* const* d_in, const int* in_sizes, int n_in,
                              void* d_out, int out_size, void* d_ws, size_t ws_size,
                              hipStream_t stream) {
    const float* feat = (const float*)d_in[0];
    const float* pos  = (const float*)d_in[1];
    const float* Wg   = (const float*)d_in[2];
    const float* bg   = (const float*)d_in[3];
    const int*   eidx = (const int*)d_in[4];
    const int*   etyp = (const int*)d_in[5];

    const int E     = in_sizes[5];          // number of edges
    const int total = out_size;             // N_NODES * C_OUT

    unsigned* keys = (unsigned*)d_out;

    init_keys<<<(total + 255) / 256, 256, 0, stream>>>(keys, total);

    const int nTiles = (E + 15) / 16;
    const int blocks = 1250;
    const int slots  = blocks * WAVES;
    const int nIter  = (nTiles + slots - 1) / slots;

    edge_wmma_kernel<<<blocks, 128, 0, stream>>>(feat, pos, Wg, bg, eidx, etyp,
                                                 keys, E, nTiles, nIter);

    finalize_keys<<<(total + 255) / 256, 256, 0, stream>>>((float*)d_out, total);
}